// Attention_76647986365039
// MI455X (gfx1250) — compile-verified
//
#include <hip/hip_runtime.h>
#include <cstddef>
#include <cstdint>

typedef float v2f __attribute__((ext_vector_type(2)));
typedef float v4f __attribute__((ext_vector_type(4)));
typedef float v8f __attribute__((ext_vector_type(8)));
typedef unsigned int v4u __attribute__((ext_vector_type(4)));
typedef int v4i __attribute__((ext_vector_type(4)));
typedef int v8i __attribute__((ext_vector_type(8)));

namespace {
constexpr int S_LEN = 2048;
constexpr int DH    = 64;
constexpr int TQ    = 128;            // q rows per workgroup (8 waves x 16)
constexpr int TKB   = 64;             // k rows per block
constexpr int NBLK  = S_LEN / TKB;    // 32
constexpr float NEG_INF_F = -1e9f;
constexpr float QK_SCALE  = 0.125f;   // 1/sqrt(64)

// LDS row strides (floats), bank-conflict-free fragment reads.
// TDM pad codes: interval code 5 => pad after every 64 DWORDs (one 64-elem row);
// amount code 3 => +4 DWORDs (stride 68), code 15 => +16 DWORDs (stride 80).
constexpr int QS_STR = 68;
constexpr int KS_STR = 68;
constexpr int VS_STR = 80;
constexpr int PS_STR = 68;
}

// Issue one 2D Tensor-Data-Mover load: global (row-major, stride0 elems of f32)
// -> LDS at lds_addr with hardware row padding. Wave-level DMA, TENSORcnt-tracked.
__device__ __forceinline__ void tdm_load_2d(unsigned lds_addr, const void* gptr,
                                            int tensor_d0, int tensor_d1,
                                            int tile_d0, int tile_d1,
                                            int stride0_elems,
                                            int pad_interval_code,
                                            int pad_amount_code) {
  unsigned long long ga = (unsigned long long)(uintptr_t)gptr;
  v4u g0;
  g0[0] = 1u;                                        // count=1, user descriptor
  g0[1] = lds_addr;                                  // lds_addr [63:32]
  g0[2] = (unsigned)(ga & 0xFFFFFFFFu);              // global_addr [95:64]
  g0[3] = (unsigned)((ga >> 32) & 0x01FFFFFFu)       // global_addr [120:96]
        | (2u << 30);                                // type=2 ("image")
  v8i g1;
  g1[0] = (2 << 16)                                  // data_size = 4 bytes
        | (1 << 20)                                  // pad_enable
        | (pad_interval_code << 22)
        | (pad_amount_code << 25);
  g1[1] = (tensor_d0 & 0xFFFF) << 16;                // tensor_dim0[15:0]
  g1[2] = ((tensor_d0 >> 16) & 0xFFFF) | ((tensor_d1 & 0xFFFF) << 16);
  g1[3] = ((tensor_d1 >> 16) & 0xFFFF) | ((tile_d0 & 0xFFFF) << 16);
  g1[4] = (tile_d1 & 0xFFFF);                        // tile_dim1, tile_dim2=0
  g1[5] = stride0_elems;                             // tensor_dim0_stride lo
  g1[6] = 0;                                         // stride0 hi, stride1 lo
  g1[7] = 0;                                         // stride1 hi
  v4i z4 = {0, 0, 0, 0};                             // groups 2/3: dims 3/4 unused
  v8i z8 = {0, 0, 0, 0, 0, 0, 0, 0};                 // trailing group (clang-23 form)
  __builtin_amdgcn_tensor_load_to_lds(g0, g1, z4, z4, z8, 0);
}

__device__ __forceinline__ unsigned lds_off(const void* p) {
  // flat address of LDS = aperture[63:32] | offset[31:0]; low 32 bits = LDS offset
  return (unsigned)(uintptr_t)p;
}

__global__ __launch_bounds__(256) void attn_fused_f32(
    const float* __restrict__ qg, const float* __restrict__ kg,
    const float* __restrict__ vg, const int* __restrict__ maskg,
    float* __restrict__ outg, float* __restrict__ pg)
{
  __shared__ float Qs[TQ * QS_STR];            // 34816 B
  __shared__ float Ks[2][TKB * KS_STR];        // 2 x 17408 B (double buffered)
  __shared__ float Vs[2][TKB * VS_STR];        // 2 x 20480 B (double buffered)
  __shared__ float Ps[8 * 16 * PS_STR];        // 34816 B per-wave transpose tiles
  __shared__ float RowScale[TQ];

  const int tid  = threadIdx.x;
  const int lane = tid & 31;
  const int wave = tid >> 5;
  const int l16  = lane & 15;   // N index (B/C/D frags) or M index (A frags)
  const int lh   = lane >> 4;   // lane half
  const int bh   = blockIdx.x;  // 0..63 (b*H + h)
  const int qt   = blockIdx.y;  // 0..15

  const float* qb = qg + (size_t)bh * S_LEN * DH + (size_t)qt * TQ * DH;
  const float* kb = kg + (size_t)bh * S_LEN * DH;
  const float* vb = vg + (size_t)bh * S_LEN * DH;
  float*       pb = pg + (size_t)bh * S_LEN * S_LEN + (size_t)qt * TQ * S_LEN;
  float*       ob = outg + (size_t)bh * S_LEN * DH + (size_t)qt * TQ * DH;
  const int*   mb = maskg + (size_t)qt * TQ * S_LEN;   // mask is [1,1,S,S]

  const unsigned qs_a  = lds_off(&Qs[0]);
  const unsigned ks_a0 = lds_off(&Ks[0][0]);
  const unsigned ks_a1 = lds_off(&Ks[1][0]);
  const unsigned vs_a0 = lds_off(&Vs[0][0]);
  const unsigned vs_a1 = lds_off(&Vs[1][0]);

  // ---- TDM prologue: Q tile (128x64) + first K block into buffer 0 ----
  if (wave == 0) {
    tdm_load_2d(qs_a, qb, DH, TQ, DH, TQ, DH, 5, 3);
    tdm_load_2d(ks_a0, kb, DH, TKB, DH, TKB, DH, 5, 3);
  }

  const int qw = wave * 16;     // wave-local q row base within tile
  float mrow[8];
#pragma unroll
  for (int vv = 0; vv < 8; ++vv) mrow[vv] = -INFINITY;

  // ================= Phase 1: exact row maxima =================
  int buf = 0;
  for (int blk = 0; blk < NBLK; ++blk) {
    if (wave == 0) __builtin_amdgcn_s_wait_tensorcnt(0);
    __syncthreads();          // current buffer DMA complete; prev reads retired
    if (wave == 0 && blk + 1 < NBLK)
      tdm_load_2d(buf ? ks_a0 : ks_a1,
                  kb + (size_t)((blk + 1) * TKB) * DH,
                  DH, TKB, DH, TKB, DH, 5, 3);      // overlaps compute below
    const float* Kb = Ks[buf];

#pragma unroll
    for (int nt = 0; nt < 4; ++nt) {
      v8f c = {0.f, 0.f, 0.f, 0.f, 0.f, 0.f, 0.f, 0.f};
#pragma unroll
      for (int ks = 0; ks < 16; ++ks) {
        v2f a = *(const v2f*)&Qs[(qw + l16) * QS_STR + 4 * ks + 2 * lh];
        v2f b = *(const v2f*)&Kb[(nt * 16 + l16) * KS_STR + 4 * ks + 2 * lh];
        c = __builtin_amdgcn_wmma_f32_16x16x4_f32(false, a, false, b,
                                                  (short)0, c, false, false);
      }
      const int kcol = blk * TKB + nt * 16 + l16;
#pragma unroll
      for (int vv = 0; vv < 8; ++vv) {
        const int qr = qw + vv + 8 * lh;
        const float s = mb[(size_t)qr * S_LEN + kcol] ? c[vv] * QK_SCALE : NEG_INF_F;
        mrow[vv] = fmaxf(mrow[vv], s);
      }
    }
    buf ^= 1;
  }
#pragma unroll
  for (int vv = 0; vv < 8; ++vv) {      // reduce across the 16 lanes of a row
    float mv = mrow[vv];
    mv = fmaxf(mv, __shfl_xor(mv, 1, 32));
    mv = fmaxf(mv, __shfl_xor(mv, 2, 32));
    mv = fmaxf(mv, __shfl_xor(mv, 4, 32));
    mv = fmaxf(mv, __shfl_xor(mv, 8, 32));
    mrow[vv] = mv;
  }

  // ================= Phase 2: exp, P write, O = P @ V =================
  v8f o[4];
#pragma unroll
  for (int nt = 0; nt < 4; ++nt) o[nt] = (v8f){0.f,0.f,0.f,0.f,0.f,0.f,0.f,0.f};
  float psum[8];
#pragma unroll
  for (int vv = 0; vv < 8; ++vv) psum[vv] = 0.f;
  float* psw = &Ps[wave * 16 * PS_STR];

  // prologue: first K+V block into buffer 0 (buffer 1 still in use by phase 1)
  if (wave == 0) {
    tdm_load_2d(ks_a0, kb, DH, TKB, DH, TKB, DH, 5, 3);
    tdm_load_2d(vs_a0, vb, DH, TKB, DH, TKB, DH, 5, 15);
  }

  buf = 0;
  for (int blk = 0; blk < NBLK; ++blk) {
    if (wave == 0) __builtin_amdgcn_s_wait_tensorcnt(0);
    __syncthreads();
    if (wave == 0 && blk + 1 < NBLK) {
      tdm_load_2d(buf ? ks_a0 : ks_a1,
                  kb + (size_t)((blk + 1) * TKB) * DH, DH, TKB, DH, TKB, DH, 5, 3);
      tdm_load_2d(buf ? vs_a0 : vs_a1,
                  vb + (size_t)((blk + 1) * TKB) * DH, DH, TKB, DH, TKB, DH, 5, 15);
    }
    const float* Kb = Ks[buf];
    const float* Vb = Vs[buf];

    // recompute scores for this K block, exponentiate, stash p
#pragma unroll
    for (int nt = 0; nt < 4; ++nt) {
      v8f c = {0.f, 0.f, 0.f, 0.f, 0.f, 0.f, 0.f, 0.f};
#pragma unroll
      for (int ks = 0; ks < 16; ++ks) {
        v2f a = *(const v2f*)&Qs[(qw + l16) * QS_STR + 4 * ks + 2 * lh];
        v2f b = *(const v2f*)&Kb[(nt * 16 + l16) * KS_STR + 4 * ks + 2 * lh];
        c = __builtin_amdgcn_wmma_f32_16x16x4_f32(false, a, false, b,
                                                  (short)0, c, false, false);
      }
      const int kcol = blk * TKB + nt * 16 + l16;
#pragma unroll
      for (int vv = 0; vv < 8; ++vv) {
        const int qr = qw + vv + 8 * lh;
        const float s = mb[(size_t)qr * S_LEN + kcol] ? c[vv] * QK_SCALE : NEG_INF_F;
        const float pe = __expf(s - mrow[vv]);   // masked -> exp(-1e9-m) == 0
        psum[vv] += pe;
        pb[(size_t)qr * S_LEN + kcol] = pe;      // unnormalized; fixed in phase 3
        psw[(vv + 8 * lh) * PS_STR + nt * 16 + l16] = pe;  // transpose staging
      }
    }

    // O += P_blk(16x64) @ V_blk(64x64)
#pragma unroll
    for (int ks = 0; ks < 16; ++ks) {
      v2f a = *(const v2f*)&psw[l16 * PS_STR + 4 * ks + 2 * lh];  // A-frag layout
      const int kr = 4 * ks + 2 * lh;
#pragma unroll
      for (int nt = 0; nt < 4; ++nt) {
        v2f b;
        b.x = Vb[kr * VS_STR + nt * 16 + l16];
        b.y = Vb[(kr + 1) * VS_STR + nt * 16 + l16];
        o[nt] = __builtin_amdgcn_wmma_f32_16x16x4_f32(false, a, false, b,
                                                      (short)0, o[nt], false, false);
      }
    }
    buf ^= 1;
  }

  // row sums -> reciprocal
#pragma unroll
  for (int vv = 0; vv < 8; ++vv) {
    float sv = psum[vv];
    sv += __shfl_xor(sv, 1, 32);
    sv += __shfl_xor(sv, 2, 32);
    sv += __shfl_xor(sv, 4, 32);
    sv += __shfl_xor(sv, 8, 32);
    psum[vv] = 1.0f / sv;
  }

  // store normalized output (C/D fragment layout)
#pragma unroll
  for (int nt = 0; nt < 4; ++nt) {
#pragma unroll
    for (int vv = 0; vv < 8; ++vv) {
      const int qr = qw + vv + 8 * lh;
      ob[(size_t)qr * DH + nt * 16 + l16] = o[nt][vv] * psum[vv];
    }
  }

  if (l16 == 0) {
#pragma unroll
    for (int vv = 0; vv < 8; ++vv) RowScale[qw + vv + 8 * lh] = psum[vv];
  }
  __syncthreads();

  // ================= Phase 3: normalize stored p rows (L2-hot) =================
  for (int r = 0; r < 16; ++r) {
    const int qr = qw + r;
    const float sc = RowScale[qr];
    float* prow = pb + (size_t)qr * S_LEN;
    for (int c4 = lane * 4; c4 < S_LEN; c4 += 128) {
      v4f t = *(v4f*)(prow + c4);
      t = t * sc;
      *(v4f*)(prow + c4) = t;
    }
  }
}

extern "C" void kernel_launch(void* const* d_in, const int* in_sizes, int n_in,
                              void* d_out, int out_size, void* d_ws, size_t ws_size,
                              hipStream_t stream) {
  (void)in_sizes; (void)n_in; (void)out_size; (void)d_ws; (void)ws_size;
  const float* q    = (const float*)d_in[0];
  const float* k    = (const float*)d_in[1];
  const float* v    = (const float*)d_in[2];
  const int*   mask = (const int*)d_in[3];

  float* out = (float*)d_out;                                   // [B,H,S,D]
  float* p   = out + (size_t)4 * 16 * S_LEN * DH;               // [B,H,S,S]

  dim3 grid(4 * 16, S_LEN / TQ);   // (bh, q-tile) = (64, 16)
  attn_fused_f32<<<grid, 256, 0, stream>>>(q, k, v, mask, out, p);
}